// RegionNormL_55765855371934
// MI455X (gfx1250) — compile-verified
//
#include <hip/hip_runtime.h>
#include <hip/hip_bf16.h>

typedef __attribute__((ext_vector_type(2))) float v2f;
typedef __attribute__((ext_vector_type(8))) float v8f;
typedef __attribute__((ext_vector_type(4))) unsigned int v4u;
typedef __attribute__((ext_vector_type(8))) int v8i;
typedef __attribute__((ext_vector_type(4))) int v4i;

constexpr int kB = 8, kH = 256, kW = 256, kC = 128;
constexpr int kHW = kH * kW;           // 65536 = 2^16
constexpr int kNPix = kB * kHW;        // 524288
constexpr float kThresh = 0.8f;
constexpr float kEps = 1e-3f;

constexpr int kTilePix = 16;                    // pixels per TDM tile
constexpr int kTileFloats = kTilePix * kC;      // 2048 floats = 8KB contiguous

// workspace layout (float offsets)
constexpr size_t OFF_PMAX = 0;
constexpr size_t OFF_PAVG = (size_t)kNPix;
constexpr size_t OFF_XS   = 2ull * kNPix;
constexpr size_t OFF_BETA = 3ull * kNPix;
constexpr size_t OFF_GAM  = 4ull * kNPix;
constexpr size_t OFF_SV1  = 5ull * kNPix;          // [B*C] sum x*flag
constexpr size_t OFF_SV2  = OFF_SV1 + kB * kC;     // [B*C] sum x^2*flag
constexpr size_t OFF_S1   = OFF_SV2 + kB * kC;     // [B*C] sum x
constexpr size_t OFF_S2   = OFF_S1  + kB * kC;     // [B*C] sum x^2
constexpr size_t OFF_CNT  = OFF_S2  + kB * kC;     // [B] flag count (16 floats pad)
constexpr size_t OFF_TMUV = OFF_CNT + 16;
constexpr size_t OFF_TAV  = OFF_TMUV + kB * kC;
constexpr size_t OFF_TMUM = OFF_TAV  + kB * kC;
constexpr size_t OFF_TAM  = OFF_TMUM + kB * kC;

// ---- Pass 1: per-pixel channel max & mean. One wave32 per pixel. ----
__global__ void pool_kernel(const float* __restrict__ in,
                            float* __restrict__ pmax, float* __restrict__ pavg) {
    const int lane = threadIdx.x & 31;
    const int gw = (blockIdx.x * blockDim.x + threadIdx.x) >> 5;
    const int nw = (gridDim.x * blockDim.x) >> 5;
    for (int p = gw; p < kNPix; p += nw) {
        const float4 v = *(const float4*)(in + (size_t)p * kC + lane * 4);
        float m = fmaxf(fmaxf(v.x, v.y), fmaxf(v.z, v.w));
        float s = (v.x + v.y) + (v.z + v.w);
        #pragma unroll
        for (int o = 16; o > 0; o >>= 1) {
            m = fmaxf(m, __shfl_xor(m, o, 32));
            s += __shfl_xor(s, o, 32);
        }
        if (lane == 0) { pmax[p] = m; pavg[p] = s * (1.0f / kC); }
    }
}

// ---- Pass 2: 3x3 conv on [max,avg] + sigmoid -> x ----
__global__ void sr_kernel(const float* __restrict__ pmax, const float* __restrict__ pavg,
                          const float* __restrict__ w_sr, const float* __restrict__ b_sr,
                          float* __restrict__ xs) {
    const int p = blockIdx.x * blockDim.x + threadIdx.x;
    if (p >= kNPix) return;
    const int b = p >> 16;
    const int r = p & (kHW - 1);
    const int h = r >> 8, w = r & 255;
    float acc = b_sr[0];
    #pragma unroll
    for (int ky = 0; ky < 3; ++ky) {
        const int hh = h + ky - 1;
        if (hh < 0 || hh >= kH) continue;
        #pragma unroll
        for (int kx = 0; kx < 3; ++kx) {
            const int ww = w + kx - 1;
            if (ww < 0 || ww >= kW) continue;
            const int n = (b << 16) + (hh << 8) + ww;
            acc = fmaf(pmax[n], w_sr[(ky * 3 + kx) * 2 + 0], acc);
            acc = fmaf(pavg[n], w_sr[(ky * 3 + kx) * 2 + 1], acc);
        }
    }
    xs[p] = 1.0f / (1.0f + __expf(-acc));
}

__global__ void zero_kernel(float* __restrict__ acc) {
    const int i = blockIdx.x * blockDim.x + threadIdx.x;
    if (i < 4 * kB * kC + 16) acc[i] = 0.0f;
}

// ---- Pass 3: beta/gamma 3x3 convs on x + per-batch mask count ----
__global__ void betagamma_kernel(const float* __restrict__ xs,
                                 const float* __restrict__ w_gamma, const float* __restrict__ b_gamma,
                                 const float* __restrict__ w_beta,  const float* __restrict__ b_beta,
                                 float* __restrict__ beta, float* __restrict__ gamma,
                                 float* __restrict__ cnt) {
    __shared__ float sm[256];
    const int p = blockIdx.x * blockDim.x + threadIdx.x;
    const int b = p >> 16;
    const int r = p & (kHW - 1);
    const int h = r >> 8, w = r & 255;
    float ab = b_beta[0], ag = b_gamma[0];
    #pragma unroll
    for (int ky = 0; ky < 3; ++ky) {
        const int hh = h + ky - 1;
        if (hh < 0 || hh >= kH) continue;
        #pragma unroll
        for (int kx = 0; kx < 3; ++kx) {
            const int ww = w + kx - 1;
            if (ww < 0 || ww >= kW) continue;
            const float xv = xs[(b << 16) + (hh << 8) + ww];
            ab = fmaf(xv, w_beta[ky * 3 + kx], ab);
            ag = fmaf(xv, w_gamma[ky * 3 + kx], ag);
        }
    }
    beta[p] = ab;
    gamma[p] = ag;
    sm[threadIdx.x] = (xs[p] > kThresh) ? 1.0f : 0.0f;
    __syncthreads();
    #pragma unroll
    for (int o = 128; o > 0; o >>= 1) {
        if (threadIdx.x < o) sm[threadIdx.x] += sm[threadIdx.x + o];
        __syncthreads();
    }
    if (threadIdx.x == 0) atomicAdd(cnt + b, sm[0]);
}

// ---- TDM: DMA one contiguous tile (kTileFloats f32) from global into LDS ----
// D# per CDNA5 ISA: group0 = {count=1, lds_addr,
// global_addr[56:0], type=2}; group1 = {data_size=4B, tensor 2048x1,
// tile 2048x1, stride 2048}. Groups 2/3/4 zero (<=2D tensor).
// This toolchain's builtin is the 6-arg flavor (g0, g1, g2, g3, g4, cpol).
__device__ __forceinline__ void tdm_load_tile(const float* gsrc, unsigned lds_off) {
    const unsigned long long ga = (unsigned long long)(uintptr_t)gsrc;
    v4u g0;
    g0[0] = 1u;                                            // count = 1 valid descriptor
    g0[1] = lds_off;                                       // lds_addr (bytes)
    g0[2] = (unsigned)ga;                                  // global_addr[31:0]
    g0[3] = (unsigned)((ga >> 32) & 0x01FFFFFFu) | (2u << 30);  // addr[56:32] | type=2
    v8i g1;
    g1[0] = (int)(2u << 16);                               // data_size = 2 (4 bytes)
    g1[1] = (int)((unsigned)(kTileFloats & 0xFFFF) << 16); // tensor_dim0[15:0]
    g1[2] = (int)((kTileFloats >> 16) | (1u << 16));       // tensor_dim0[31:16] | tensor_dim1=1
    g1[3] = (int)((unsigned)(kTileFloats & 0xFFFF) << 16); // tile_dim0 = 2048
    g1[4] = 1;                                             // tile_dim1 = 1 (tile_dim2 = 0)
    g1[5] = kTileFloats;                                   // tensor_dim0_stride[31:0]
    g1[6] = 0;
    g1[7] = 0;
    const v4i gz4 = {};
    const v8i gz8 = {};
    __builtin_amdgcn_tensor_load_to_lds(g0, g1, gz4, gz4, gz8, 0);
}

// ---- Pass 4: per-(b,c) moments. TDM double-buffers 8KB pixel tiles into LDS;
// 8 waves consume 16-channel slices as V_WMMA_F32_16X16X4_F32 A-fragments.
// B col0 = flag, col1 = 1.0 -> masked and total sums from one WMMA chain;
// second chain on A*A gives second moments. All exact f32.
__global__ void moments_kernel(const float* __restrict__ in, const float* __restrict__ xs,
                               float* __restrict__ sv1, float* __restrict__ sv2,
                               float* __restrict__ s1,  float* __restrict__ s2) {
    __shared__ float lbuf[2][kTileFloats];
    const int b = blockIdx.y;
    const int lane = threadIdx.x & 31;
    const int wid = threadIdx.x >> 5;
    const int c_base = wid << 4;          // 8 waves cover all 128 channels
    const int row = lane & 15;            // A: M index / B,D: N index
    const int hi = lane >> 4;
    const float* in_b = in + (size_t)b * kHW * kC;
    const float* xs_b = xs + (size_t)b * kHW;
    const int tiles_per_block = (kHW / kTilePix) / gridDim.x;
    const int t0 = blockIdx.x * tiles_per_block;
    const int t_end = t0 + tiles_per_block;
    const unsigned lds0 = (unsigned)(uintptr_t)&lbuf[0][0];
    const unsigned lds1 = (unsigned)(uintptr_t)&lbuf[1][0];

    v8f acc1 = {};
    v8f acc2 = {};

    if (wid == 0) {
        tdm_load_tile(in_b + (size_t)t0 * kTileFloats, lds0);
        __builtin_amdgcn_s_wait_tensorcnt(0);
    }
    __syncthreads();

    int cur = 0;
    for (int t = t0; t < t_end; ++t) {
        if (wid == 0 && (t + 1) < t_end) {  // prefetch next tile into other buffer
            tdm_load_tile(in_b + (size_t)(t + 1) * kTileFloats, cur ? lds0 : lds1);
        }
        const float* lb = &lbuf[cur][0];
        const int pbase = t * kTilePix;
        #pragma unroll
        for (int sub = 0; sub < 4; ++sub) {
            const int k0 = sub * 4 + hi * 2;  // A v0: K=0(lo)/K=2(hi); v1: K=1/K=3
            const float* lp = lb + k0 * kC + c_base + row;
            v2f a;
            a.x = lp[0];
            a.y = lp[kC];
            const float f0 = (xs_b[pbase + k0] > kThresh) ? 1.0f : 0.0f;
            const float f1 = (xs_b[pbase + k0 + 1] > kThresh) ? 1.0f : 0.0f;
            v2f bm;
            bm.x = (row == 0) ? f0 : ((row == 1) ? 1.0f : 0.0f);
            bm.y = (row == 0) ? f1 : ((row == 1) ? 1.0f : 0.0f);
            v2f a2;
            a2.x = a.x * a.x;
            a2.y = a.y * a.y;
            acc1 = __builtin_amdgcn_wmma_f32_16x16x4_f32(false, a, false, bm, (short)0,
                                                         acc1, false, false);
            acc2 = __builtin_amdgcn_wmma_f32_16x16x4_f32(false, a2, false, bm, (short)0,
                                                         acc2, false, false);
        }
        if (wid == 0) __builtin_amdgcn_s_wait_tensorcnt(0);  // prefetch landed
        __syncthreads();  // consumers done with cur; next buffer ready
        cur ^= 1;
    }

    // D layout: VGPR r -> M=r (lanes 0-15) / M=8+r (lanes 16-31), N = lane%16.
    if (row < 2) {
        const int cb = b * kC + c_base + hi * 8;
        float* d1 = (row == 0 ? sv1 : s1) + cb;
        float* d2 = (row == 0 ? sv2 : s2) + cb;
        #pragma unroll
        for (int r2 = 0; r2 < 8; ++r2) {
            atomicAdd(d1 + r2, acc1[r2]);
            atomicAdd(d2 + r2, acc2[r2]);
        }
    }
}

// ---- Pass 5: turn sums into (mu, rsqrt(var+eps)) tables for both regions ----
__global__ void finalize_kernel(const float* __restrict__ sv1, const float* __restrict__ sv2,
                                const float* __restrict__ s1,  const float* __restrict__ s2,
                                const float* __restrict__ cnt,
                                float* __restrict__ tmuv, float* __restrict__ tav,
                                float* __restrict__ tmum, float* __restrict__ tam) {
    const int i = blockIdx.x * blockDim.x + threadIdx.x;
    if (i >= kB * kC) return;
    const int b = i >> 7;
    const float cv = cnt[b];
    const float fsv = cv + kEps;
    const float a1 = sv1[i], a2 = sv2[i];
    const float m1 = a1 / fsv;
    const float var_v = (a2 - 2.0f * m1 * a1 + m1 * m1 * cv) / fsv;
    const float cm = (float)kHW - cv;
    const float fsm = cm + kEps;
    const float b1 = s1[i] - a1, b2 = s2[i] - a2;
    const float m2 = b1 / fsm;
    const float var_m = (b2 - 2.0f * m2 * b1 + m2 * m2 * cm) / fsm;
    tmuv[i] = m1;
    tav[i]  = rsqrtf(var_v + kEps);
    tmum[i] = m2;
    tam[i]  = rsqrtf(var_m + kEps);
}

// ---- Pass 6: normalize + affine. One wave32 per pixel, float4 per lane. ----
__global__ void output_kernel(const float* __restrict__ in, const float* __restrict__ xs,
                              const float* __restrict__ beta, const float* __restrict__ gamma,
                              const float* __restrict__ tmuv, const float* __restrict__ tav,
                              const float* __restrict__ tmum, const float* __restrict__ tam,
                              float* __restrict__ out) {
    const int lane = threadIdx.x & 31;
    const int gw = (blockIdx.x * blockDim.x + threadIdx.x) >> 5;
    const int nw = (gridDim.x * blockDim.x) >> 5;
    for (int p = gw; p < kNPix; p += nw) {
        const int b = p >> 16;
        const int c0 = lane * 4;
        const size_t base = (size_t)p * kC + c0;
        const float4 xv = *(const float4*)(in + base);
        __builtin_prefetch((const void*)(in + base + (size_t)nw * kC), 0, 0);
        const bool fl = xs[p] > kThresh;
        const float bt = beta[p], gm = gamma[p];
        const int ti = b * kC + c0;
        const float4 mu = fl ? *(const float4*)(tmuv + ti) : *(const float4*)(tmum + ti);
        const float4 aa = fl ? *(const float4*)(tav + ti)  : *(const float4*)(tam + ti);
        float4 o;
        o.x = (xv.x - mu.x) * aa.x * bt + gm;
        o.y = (xv.y - mu.y) * aa.y * bt + gm;
        o.z = (xv.z - mu.z) * aa.z * bt + gm;
        o.w = (xv.w - mu.w) * aa.w * bt + gm;
        *(float4*)(out + base) = o;
    }
}

extern "C" void kernel_launch(void* const* d_in, const int* in_sizes, int n_in,
                              void* d_out, int out_size, void* d_ws, size_t ws_size,
                              hipStream_t stream) {
    const float* in      = (const float*)d_in[0];
    const float* w_sr    = (const float*)d_in[1];
    const float* b_sr    = (const float*)d_in[2];
    const float* w_gamma = (const float*)d_in[3];
    const float* b_gamma = (const float*)d_in[4];
    const float* w_beta  = (const float*)d_in[5];
    const float* b_beta  = (const float*)d_in[6];
    float* ws  = (float*)d_ws;
    float* out = (float*)d_out;

    float* pmax  = ws + OFF_PMAX;
    float* pavg  = ws + OFF_PAVG;
    float* xs    = ws + OFF_XS;
    float* beta  = ws + OFF_BETA;
    float* gamma = ws + OFF_GAM;
    float* sv1   = ws + OFF_SV1;
    float* sv2   = ws + OFF_SV2;
    float* s1    = ws + OFF_S1;
    float* s2    = ws + OFF_S2;
    float* cnt   = ws + OFF_CNT;
    float* tmuv  = ws + OFF_TMUV;
    float* tav   = ws + OFF_TAV;
    float* tmum  = ws + OFF_TMUM;
    float* tam   = ws + OFF_TAM;

    pool_kernel<<<2048, 256, 0, stream>>>(in, pmax, pavg);
    sr_kernel<<<kNPix / 256, 256, 0, stream>>>(pmax, pavg, w_sr, b_sr, xs);
    zero_kernel<<<17, 256, 0, stream>>>(ws + OFF_SV1);
    betagamma_kernel<<<kNPix / 256, 256, 0, stream>>>(xs, w_gamma, b_gamma, w_beta, b_beta,
                                                      beta, gamma, cnt);
    moments_kernel<<<dim3(128, kB), 256, 0, stream>>>(in, xs, sv1, sv2, s1, s2);
    finalize_kernel<<<(kB * kC + 255) / 256, 256, 0, stream>>>(sv1, sv2, s1, s2, cnt,
                                                               tmuv, tav, tmum, tam);
    output_kernel<<<2048, 256, 0, stream>>>(in, xs, beta, gamma, tmuv, tav, tmum, tam, out);
}